// CriterionPairWiseforWholeFeatAfterPool_28183575396881
// MI455X (gfx1250) — compile-verified
//
#include <hip/hip_runtime.h>
#include <math.h>

typedef __attribute__((ext_vector_type(4))) float f4;
typedef __attribute__((ext_vector_type(2))) float v2f;
typedef __attribute__((ext_vector_type(8))) float v8f;

// Tensor geometry (from reference): preds[feat_ind] -> (B=16, C=256, H=128, W=128)
#define NB 16
#define NC 256
#define HW 128
#define IMG (HW * HW)          // 16384 floats per (b,c) image
#define FEAT_ELEMS (16LL * 256LL * 128LL * 128LL)

// ---------------------------------------------------------------------------
// Kernel 1: streaming quadrant max-pool.
// One block per (tensor, b, c, quadrant). Block reduces a 64x64 quadrant
// (16 KB) with float4 nontemporal loads -> one pooled float in workspace.
// pooled layout: [((t*16+b)*256+c)*4 + q]  == blockIdx.x
// ---------------------------------------------------------------------------
__global__ __launch_bounds__(256) void pool_quadrant_max(
    const float* __restrict__ S, const float* __restrict__ T,
    const int* __restrict__ feat_ind, float* __restrict__ pooled) {
  const int bid = blockIdx.x;
  const int q = bid & 3;            // quadrant: m = 2*qi + qj (row-major 2x2)
  const int c = (bid >> 2) & 255;
  const int b = (bid >> 10) & 15;
  const int t = bid >> 14;          // 0 = S, 1 = T

  const float* src = (t == 0) ? S : T;
  const long long fofs = (long long)feat_ind[0] * FEAT_ELEMS;
  const float* img = src + fofs + ((long long)(b * NC + c)) * IMG;
  const int qi = q >> 1, qj = q & 1;
  const float* quad = img + qi * 64 * HW + qj * 64;

  const int tid = threadIdx.x;
  float m = -INFINITY;
  // Quadrant = 64 rows x 16 float4 slots = 1024 slots; 256 threads x 4 iters.
#pragma unroll
  for (int it = 0; it < 4; ++it) {
    const int s = it * 256 + tid;
    const int row = s >> 4;
    const int col4 = s & 15;
    const f4* p = (const f4*)(quad + row * HW + col4 * 4);
    f4 v = __builtin_nontemporal_load(p);
    m = fmaxf(m, fmaxf(fmaxf(v.x, v.y), fmaxf(v.z, v.w)));
  }

  __shared__ float sm[256];
  sm[tid] = m;
  __syncthreads();
#pragma unroll
  for (int off = 128; off > 0; off >>= 1) {
    if (tid < off) sm[tid] = fmaxf(sm[tid], sm[tid + off]);
    __syncthreads();
  }
  if (tid == 0) pooled[bid] = sm[0];
}

// ---------------------------------------------------------------------------
// Kernel 2: one wave. Norms -> WMMA grams (4 batches block-diagonal per
// 16x16 tile, K=256 in chunks of 4 via v_wmma_f32_16x16x4_f32) -> loss.
// ---------------------------------------------------------------------------
__global__ __launch_bounds__(32) void gram_loss(
    const float* __restrict__ pooled, float* __restrict__ out) {
  __shared__ float norms[2 * NB * 4];   // [t][b][m]
  __shared__ float dmat[8 * 256];       // 8 tiles (t*4+g), each 16x16

  const int lane = threadIdx.x;

  // ---- channel-L2 norms: 128 of them, 4 per lane -------------------------
  for (int i = lane; i < 128; i += 32) {
    const int t = i >> 6;
    const int b = (i >> 2) & 15;
    const int m = i & 3;
    const float* p = pooled + (long long)((t * NB + b) * NC) * 4 + m;
    float s = 0.f;
    for (int cc = 0; cc < NC; ++cc) {
      const float v = p[cc * 4];
      s += v * v;
    }
    norms[i] = sqrtf(s) + 1e-8f;
  }
  __syncthreads();

  // ---- WMMA gram: A == B (symmetric packing) -----------------------------
  // Lane mapping (both A 16x4 and B 4x16 f32 layouts):
  //   row/col index = lane&15 -> local batch bb = (lane&15)>>2, m = lane&3
  //   lanes 0-15 carry K = {k0, k0+1}; lanes 16-31 carry K = {k0+2, k0+3}
  const int bb = (lane & 15) >> 2;
  const int mm = lane & 3;
  const int klo = (lane >> 4) << 1;
  const int N = lane & 15;
  const int Mofs = (lane >> 4) * 8;

  for (int t = 0; t < 2; ++t) {
    for (int g = 0; g < 4; ++g) {
      const int bat = g * 4 + bb;
      const float inv = 1.0f / norms[t * 64 + bat * 4 + mm];
      const float* p = pooled + (long long)((t * NB + bat) * NC) * 4 + mm;
      v8f acc = {};
      for (int c0 = 0; c0 < NC; c0 += 4) {
        v2f a;
        a.x = p[(c0 + klo) * 4] * inv;
        a.y = p[(c0 + klo + 1) * 4] * inv;
        // D = A * A^T + acc  (block-diagonal 4x4 grams of 4 batches)
        acc = __builtin_amdgcn_wmma_f32_16x16x4_f32(
            false, a, false, a, (short)0, acc, false, false);
      }
      // D layout: VGPR r, lanes 0-15 -> (M=r, N=lane); lanes 16-31 -> M=r+8
      const int base = (t * 4 + g) * 256;
#pragma unroll
      for (int r = 0; r < 8; ++r) dmat[base + (Mofs + r) * 16 + N] = acc[r];
    }
  }
  __syncthreads();

  // ---- loss: diagonal 4x4 blocks only, (G_T - G_S)^2 / (M*M) / B ---------
  float s = 0.f;
  for (int i = lane; i < 4 * 256; i += 32) {
    const int g = i >> 8;
    const int e = i & 255;
    const int M = e >> 4;
    const int Nn = e & 15;
    if ((M >> 2) == (Nn >> 2)) {                  // same-batch block
      const float d = dmat[(4 + g) * 256 + e]      // tensor T
                    - dmat[g * 256 + e];           // tensor S
      s += d * d;
    }
  }
  s *= (1.0f / 256.0f);   // /(M*M)=16, /B=16
#pragma unroll
  for (int off = 16; off > 0; off >>= 1) s += __shfl_xor(s, off, 32);
  if (lane == 0) out[0] = s;
}

extern "C" void kernel_launch(void* const* d_in, const int* in_sizes, int n_in,
                              void* d_out, int out_size, void* d_ws, size_t ws_size,
                              hipStream_t stream) {
  const float* preds_S = (const float*)d_in[0];
  const float* preds_T = (const float*)d_in[1];
  const int* feat_ind  = (const int*)d_in[2];
  float* pooled = (float*)d_ws;            // 2*16*256*4 floats = 128 KB
  float* out = (float*)d_out;

  // Stage 1: 2 tensors * 16 b * 256 c * 4 quadrants = 32768 blocks.
  pool_quadrant_max<<<dim3(2 * NB * NC * 4), dim3(256), 0, stream>>>(
      preds_S, preds_T, feat_ind, pooled);

  // Stage 2: single wave does norms + WMMA grams + loss.
  gram_loss<<<dim3(1), dim3(32), 0, stream>>>(pooled, out);
}